// Sampler_13426067768042
// MI455X (gfx1250) — compile-verified
//
#include <hip/hip_runtime.h>
#include <cstdint>
#include <cfloat>
#include <climits>

// ---------------- CDNA5-specific async streaming primitives ----------------
// Per-lane async copy of 16B global -> LDS (GLOBAL_LOAD_ASYNC_TO_LDS_B128,
// GV addressing mode, tracked by ASYNCcnt). EXEC-masked per lane.
__device__ __forceinline__ void async_ld16(unsigned lds_byte, const void* g) {
  asm volatile("global_load_async_to_lds_b128 %0, %1, off"
               :: "v"(lds_byte), "v"((unsigned long long)(uintptr_t)g)
               : "memory");
}
#define WAIT_ASYNC0() asm volatile("s_wait_asynccnt 0" ::: "memory")
#define WAIT_ASYNC1() asm volatile("s_wait_asynccnt 1" ::: "memory")
#define WAIT_ASYNC2() asm volatile("s_wait_asynccnt 2" ::: "memory")

// Order-preserving float->uint key (ascending), and bin lower-edge inverse.
__device__ __forceinline__ unsigned fkey(float x) {
  unsigned u = __float_as_uint(x);
  return (u & 0x80000000u) ? ~u : (u | 0x80000000u);
}
__device__ __forceinline__ float bin_lo(int b) {
  unsigned u = ((unsigned)b) << 20;
  return (u & 0x80000000u) ? __uint_as_float(u & 0x7fffffffu)
                           : __uint_as_float(~u);
}

constexpr int V_SZ  = 128000;
constexpr int BLK   = 640;          // 20 wave32
constexpr int TILE  = BLK * 4;      // 2560 floats / tile, 10KB
constexpr int NT    = V_SZ / TILE;  // 50 tiles, exact
constexpr int NWAVE = BLK / 32;     // 20
constexpr int NBIN  = 4096;
constexpr int SCANT = 512;
constexpr int BPT   = NBIN / SCANT; // 8 bins per scan thread

__global__ __launch_bounds__(BLK) void sampler_kernel(
    const float* __restrict__ logits, const float* __restrict__ temps,
    const int*   __restrict__ topk,   const float* __restrict__ topp,
    const float* __restrict__ minp,   const float* __restrict__ noise,
    float* __restrict__ out)
{
  __shared__ __align__(16) float s_lg[2][TILE];
  __shared__ __align__(16) float s_ns[2][TILE];
  __shared__ unsigned histC[NBIN];
  __shared__ float    histM[NBIN];
  __shared__ unsigned s_scanC[SCANT];
  __shared__ float    s_scanM[SCANT];
  __shared__ float swm[NWAVE], sws[NWAVE], swb[NWAVE];
  __shared__ int   swi[NWAVE];
  __shared__ float sh_M, sh_Z, sh_T;
  __shared__ int   sh_greedy, sh_dk, sh_dp;

  const int row  = blockIdx.x;
  const int tid  = threadIdx.x;
  const int lane = tid & 31, wv = tid >> 5;

  const float temp  = temps[row];
  const float tempc = fmaxf(temp, 1e-10f);
  const float invT  = 1.0f / tempc;
  const int   k     = topk[row];
  const float tp    = topp[row];
  const float mp    = minp[row];

  const float* rowL = logits + (size_t)row * V_SZ;
  const float* rowN = noise  + (size_t)row * V_SZ;

  const unsigned ldsLG = (unsigned)(uintptr_t)&s_lg[0][0]; // LDS offset = low 32b
  const unsigned ldsNS = (unsigned)(uintptr_t)&s_ns[0][0];

  for (int i = tid; i < NBIN; i += BLK) { histC[i] = 0u; histM[i] = 0.f; }

  // ---------------- Pass 1: online softmax + greedy argmax ----------------
  float m = -FLT_MAX, s = 0.f, gb = -FLT_MAX;
  int gi = INT_MAX;
  async_ld16(ldsLG + (unsigned)(tid * 16), rowL + tid * 4);
  for (int t = 0; t < NT; ++t) {
    const int p = t & 1;
    if (t + 1 < NT) {
      const int q = (t + 1) & 1;
      async_ld16(ldsLG + (unsigned)((q * TILE + tid * 4) * 4),
                 rowL + (t + 1) * TILE + tid * 4);
      WAIT_ASYNC1();
    } else WAIT_ASYNC0();
    const float4 v = *(const float4*)&s_lg[p][tid * 4];
    const int base = t * TILE + tid * 4;
    float xs[4] = {v.x, v.y, v.z, v.w};
    #pragma unroll
    for (int j = 0; j < 4; ++j) {
      float x = xs[j];
      if (x > gb) { gb = x; gi = base + j; }
      if (x > m)  { s = s * __expf((m - x) * invT) + 1.f; m = x; }
      else        { s += __expf((x - m) * invT); }
    }
  }
  #pragma unroll
  for (int off = 16; off > 0; off >>= 1) {
    float om = __shfl_down(m, off);
    float os = __shfl_down(s, off);
    float nm = fmaxf(m, om);
    s = s * __expf((m - nm) * invT) + os * __expf((om - nm) * invT);
    m = nm;
    float ob = __shfl_down(gb, off);
    int   oi = __shfl_down(gi, off);
    if (ob > gb || (ob == gb && oi < gi)) { gb = ob; gi = oi; }
  }
  if (lane == 0) { swm[wv] = m; sws[wv] = s; swb[wv] = gb; swi[wv] = gi; }
  __syncthreads();
  if (tid == 0) {
    float M = swm[0], S = sws[0], bb = swb[0]; int bi = swi[0];
    for (int w = 1; w < NWAVE; ++w) {
      float nm = fmaxf(M, swm[w]);
      S = S * __expf((M - nm) * invT) + sws[w] * __expf((swm[w] - nm) * invT);
      M = nm;
      if (swb[w] > bb || (swb[w] == bb && swi[w] < bi)) { bb = swb[w]; bi = swi[w]; }
    }
    sh_M = M; sh_Z = S; sh_greedy = bi;
    sh_dk = NBIN; sh_dp = NBIN;
  }
  __syncthreads();

  if (temp == 0.0f) {                 // greedy rows: done
    if (tid == 0) out[row] = (float)sh_greedy;
    return;
  }

  const float M = sh_M, Z = sh_Z;
  const float Tminp = M + tempc * __logf(mp);   // min_p as a logit threshold

  // ---------------- Pass 2 (L2-resident): count + mass histogram ----------
  async_ld16(ldsLG + (unsigned)(tid * 16), rowL + tid * 4);
  for (int t = 0; t < NT; ++t) {
    const int p = t & 1;
    if (t + 1 < NT) {
      const int q = (t + 1) & 1;
      async_ld16(ldsLG + (unsigned)((q * TILE + tid * 4) * 4),
                 rowL + (t + 1) * TILE + tid * 4);
      WAIT_ASYNC1();
    } else WAIT_ASYNC0();
    const float4 v = *(const float4*)&s_lg[p][tid * 4];
    float xs[4] = {v.x, v.y, v.z, v.w};
    #pragma unroll
    for (int j = 0; j < 4; ++j) {
      float x = xs[j];
      if (x >= Tminp) {
        unsigned b = fkey(x) >> 20;
        atomicAdd(&histC[b], 1u);                       // ds_add_u32
        atomicAdd(&histM[b], __expf((x - M) * invT));   // ds_add_f32
      }
    }
  }
  __syncthreads();

  // ---------------- Selection scan over 4096 bins (descending) ------------
  const bool kActive = (k > 0) && (k < V_SZ);
  const bool pActive = (tp < 1.0f);
  if (tid < SCANT) {
    unsigned c = 0;
    #pragma unroll
    for (int j = 0; j < BPT; ++j) c += histC[NBIN - 1 - (tid * BPT + j)];
    s_scanC[tid] = c;
  }
  __syncthreads();
  if (tid == 0) {
    unsigned run = 0;
    for (int t = 0; t < SCANT; ++t) { unsigned c = s_scanC[t]; s_scanC[t] = run; run += c; }
  }
  __syncthreads();
  if (tid < SCANT && kActive) {
    unsigned cum = s_scanC[tid];
    #pragma unroll
    for (int j = 0; j < BPT; ++j) {
      int d = tid * BPT + j;
      cum += histC[NBIN - 1 - d];
      if (cum >= (unsigned)k) { atomicMin(&sh_dk, d); break; }
    }
  }
  __syncthreads();
  const int dk = sh_dk;               // NBIN => no k-cutoff
  if (tid < SCANT) {
    float msum = 0.f;
    #pragma unroll
    for (int j = 0; j < BPT; ++j) {
      int d = tid * BPT + j;
      if (d <= dk) msum += histM[NBIN - 1 - d];   // zero mass past k-cutoff
    }
    s_scanM[tid] = msum;
  }
  __syncthreads();
  if (tid == 0) {
    float run = 0.f;
    for (int t = 0; t < SCANT; ++t) { float c = s_scanM[t]; s_scanM[t] = run; run += c; }
  }
  __syncthreads();
  const float tpZ = tp * Z;           // compare unnormalized mass vs top_p*Z
  if (tid < SCANT && pActive) {
    float cum = s_scanM[tid];
    #pragma unroll
    for (int j = 0; j < BPT; ++j) {
      int d = tid * BPT + j;
      if (d <= dk) cum += histM[NBIN - 1 - d];
      if (cum > tpZ) { atomicMin(&sh_dp, d); break; }
    }
  }
  __syncthreads();
  if (tid == 0) {
    float T = Tminp;
    if (kActive && sh_dk < NBIN) T = fmaxf(T, bin_lo(NBIN - 1 - sh_dk));
    if (pActive && sh_dp < NBIN) T = fmaxf(T, bin_lo(NBIN - 1 - sh_dp));
    sh_T = T;
  }
  __syncthreads();
  const float T = sh_T;

  // ---------------- Pass 3: Gumbel-max over kept set (logits + noise) -----
  float best = -FLT_MAX; int bidx = INT_MAX;
  async_ld16(ldsLG + (unsigned)(tid * 16), rowL + tid * 4);
  async_ld16(ldsNS + (unsigned)(tid * 16), rowN + tid * 4);
  for (int t = 0; t < NT; ++t) {
    const int p = t & 1;
    if (t + 1 < NT) {
      const int q = (t + 1) & 1;
      async_ld16(ldsLG + (unsigned)((q * TILE + tid * 4) * 4),
                 rowL + (t + 1) * TILE + tid * 4);
      async_ld16(ldsNS + (unsigned)((q * TILE + tid * 4) * 4),
                 rowN + (t + 1) * TILE + tid * 4);
      WAIT_ASYNC2();
    } else WAIT_ASYNC0();
    const float4 lv = *(const float4*)&s_lg[p][tid * 4];
    const float4 nv = *(const float4*)&s_ns[p][tid * 4];
    const int base = t * TILE + tid * 4;
    float lx[4] = {lv.x, lv.y, lv.z, lv.w};
    float nx[4] = {nv.x, nv.y, nv.z, nv.w};
    #pragma unroll
    for (int j = 0; j < 4; ++j) {
      float x = lx[j];
      if (x >= T) {
        float r = (x - M) * invT - __logf(nx[j] + 1e-10f);  // log-space p/e
        int idx = base + j;
        if (r > best || (r == best && idx < bidx)) { best = r; bidx = idx; }
      }
    }
  }
  #pragma unroll
  for (int off = 16; off > 0; off >>= 1) {
    float ov = __shfl_down(best, off);
    int   oi = __shfl_down(bidx, off);
    if (ov > best || (ov == best && oi < bidx)) { best = ov; bidx = oi; }
  }
  if (lane == 0) { swb[wv] = best; swi[wv] = bidx; }
  __syncthreads();
  if (tid == 0) {
    float bb = swb[0]; int bi = swi[0];
    for (int w = 1; w < NWAVE; ++w)
      if (swb[w] > bb || (swb[w] == bb && swi[w] < bi)) { bb = swb[w]; bi = swi[w]; }
    out[row] = (float)bi;
  }
}

extern "C" void kernel_launch(void* const* d_in, const int* in_sizes, int n_in,
                              void* d_out, int out_size, void* d_ws, size_t ws_size,
                              hipStream_t stream) {
  (void)n_in; (void)out_size; (void)d_ws; (void)ws_size;
  const float* logits = (const float*)d_in[0];
  const float* temps  = (const float*)d_in[1];
  const int*   topk   = (const int*)d_in[2];
  const float* topp   = (const float*)d_in[3];
  const float* minp   = (const float*)d_in[4];
  const float* noise  = (const float*)d_in[5];
  float* out = (float*)d_out;
  const int B = in_sizes[1];   // one workgroup per row
  hipLaunchKernelGGL(sampler_kernel, dim3(B), dim3(BLK), 0, stream,
                     logits, temps, topk, topp, minp, noise, out);
}